// SelectiveSSM_73040213835811
// MI455X (gfx1250) — compile-verified
//
#include <hip/hip_runtime.h>
#include <hip/hip_bf16.h>

typedef __attribute__((ext_vector_type(2))) float v2f;
typedef __attribute__((ext_vector_type(8))) float v8f;

#define D_INNER 1024
#define STATE   16
#define DT_RANK 64
#define BATCH   2
#define SEQ     2048
#define ROWS    (BATCH * SEQ)            // 4096
#define E_DIM   (DT_RANK + 2 * STATE)    // 96
#define NCHUNK  8
#define CLEN    (SEQ / NCHUNK)           // 256

// ---------------------------------------------------------------------------
// Kernel 1: x_dbl[r][e] = sum_d u[r][d] * W_x[e][d]   (4096x96 = 4096x1024 * 1024x96)
// One wave per 16x16 output tile, K-loop of V_WMMA_F32_16X16X4_F32 (full fp32).
// ---------------------------------------------------------------------------
__global__ __launch_bounds__(128) void xproj_wmma(
    const float* __restrict__ u, const float* __restrict__ Wx,
    float* __restrict__ x_dbl) {
  const int wave = blockIdx.x * (blockDim.x >> 5) + (threadIdx.x >> 5);
  const int lane = threadIdx.x & 31;
  const int rt = wave / (E_DIM / 16);     // 0..255
  const int ct = wave % (E_DIM / 16);     // 0..5
  const int row0 = rt * 16, col0 = ct * 16;
  const int m  = lane & 15;               // M (A) / N (B) index
  const int kh = lane >> 4;               // K half-select

  v8f acc = {};
  const float* ua = u  + (size_t)(row0 + m) * D_INNER + 2 * kh;
  const float* wa = Wx + (size_t)(col0 + m) * D_INNER + 2 * kh;  // B[k][n] = Wx[n][k]
#pragma unroll 8
  for (int kb = 0; kb < D_INNER; kb += 4) {
    v2f a = *(const v2f*)(ua + kb);   // A[m][2*kh + j]
    v2f b = *(const v2f*)(wa + kb);   // B[2*kh + j][n]
    acc = __builtin_amdgcn_wmma_f32_16x16x4_f32(
        /*neg_a=*/false, a, /*neg_b=*/false, b,
        /*c_mod=*/(short)0, acc, /*reuse_a=*/false, /*reuse_b=*/false);
  }
  // C/D layout: VGPR j -> row (j + 8*kh), col m
  float* outp = x_dbl + (size_t)(row0 + 8 * kh) * E_DIM + col0 + m;
#pragma unroll
  for (int j = 0; j < 8; ++j) outp[(size_t)j * E_DIM] = acc[j];
}

// ---------------------------------------------------------------------------
// Kernel 2: dt[r][d] = softplus( sum_k x_dbl[r][k] * W_dt[d][k] + b_dt[d] )
// (4096x1024 = 4096x64 * 64x1024), K=64 -> 16 WMMAs per tile.
// ---------------------------------------------------------------------------
__global__ __launch_bounds__(128) void dtproj_wmma(
    const float* __restrict__ x_dbl, const float* __restrict__ Wdt,
    const float* __restrict__ b_dt, float* __restrict__ dt) {
  const int wave = blockIdx.x * (blockDim.x >> 5) + (threadIdx.x >> 5);
  const int lane = threadIdx.x & 31;
  const int rt = wave / (D_INNER / 16);   // 0..255
  const int ct = wave % (D_INNER / 16);   // 0..63
  const int row0 = rt * 16, col0 = ct * 16;
  const int m  = lane & 15;
  const int kh = lane >> 4;

  v8f acc = {};
  const float* aa = x_dbl + (size_t)(row0 + m) * E_DIM + 2 * kh;  // first 64 cols = dt_raw
  const float* wa = Wdt   + (size_t)(col0 + m) * DT_RANK + 2 * kh;
#pragma unroll
  for (int kb = 0; kb < DT_RANK; kb += 4) {
    v2f a = *(const v2f*)(aa + kb);
    v2f b = *(const v2f*)(wa + kb);
    acc = __builtin_amdgcn_wmma_f32_16x16x4_f32(
        false, a, false, b, (short)0, acc, false, false);
  }
  const float bias = b_dt[col0 + m];
  float* outp = dt + (size_t)(row0 + 8 * kh) * D_INNER + col0 + m;
#pragma unroll
  for (int j = 0; j < 8; ++j) {
    float x = acc[j] + bias;
    float sp = (x > 20.0f) ? x : log1pf(__expf(x));   // softplus
    outp[(size_t)j * D_INNER] = sp;
  }
}

// ---------------------------------------------------------------------------
// Scan pass 1: per-chunk summaries. One wave = 2 d-channels x 16 states.
// For its chunk each lane computes prod(A_bar) and h at chunk end (h_in = 0).
// ---------------------------------------------------------------------------
__global__ __launch_bounds__(256) void scan_pass1(
    const float* __restrict__ u, const float* __restrict__ A_log,
    const float* __restrict__ x_dbl, const float* __restrict__ dt,
    float* __restrict__ prodA, float* __restrict__ hend) {
  const int wave = blockIdx.x * (blockDim.x >> 5) + (threadIdx.x >> 5);
  const int lane = threadIdx.x & 31;
  const int b     = wave / (NCHUNK * (D_INNER / 2));
  const int rem   = wave % (NCHUNK * (D_INNER / 2));
  const int chunk = rem / (D_INNER / 2);
  const int dpair = rem % (D_INNER / 2);
  const int d = dpair * 2 + (lane >> 4);
  const int n = lane & 15;

  const float A = -__expf(A_log[d * STATE + n]);
  float h = 0.0f, p = 1.0f;
  const int lbase = chunk * CLEN;
  for (int i = 0; i < CLEN; ++i) {
    const size_t r = (size_t)b * SEQ + (lbase + i);
    const float dtv = dt[r * D_INNER + d];
    const float uv  = u [r * D_INNER + d];
    const float Bv  = x_dbl[r * E_DIM + DT_RANK + n];
    const float a = __expf(dtv * A);
    h = fmaf(a, h, dtv * uv * Bv);
    p *= a;
  }
  const size_t sidx = (((size_t)b * D_INNER + d) * STATE + n) * NCHUNK + chunk;
  prodA[sidx] = p;
  hend[sidx]  = h;
}

// ---------------------------------------------------------------------------
// Scan pass 2: serial combine over the 8 chunk summaries -> per-chunk carry-in.
// One thread per (b,d,n).
// ---------------------------------------------------------------------------
__global__ __launch_bounds__(256) void scan_combine(
    const float* __restrict__ prodA, const float* __restrict__ hend,
    float* __restrict__ hin) {
  const int t = blockIdx.x * blockDim.x + threadIdx.x;   // 0..32767
  const size_t base = (size_t)t * NCHUNK;
  float h = 0.0f;
#pragma unroll
  for (int c = 0; c < NCHUNK; ++c) {
    hin[base + c] = h;
    h = fmaf(prodA[base + c], h, hend[base + c]);
  }
}

// ---------------------------------------------------------------------------
// Scan pass 3: replay chunk with carry-in, contract over N via shfl_xor
// (wave32: xor masks 1,2,4,8 stay within each 16-lane state group),
// write y + u*D_param.
// ---------------------------------------------------------------------------
__global__ __launch_bounds__(256) void scan_pass3(
    const float* __restrict__ u, const float* __restrict__ A_log,
    const float* __restrict__ x_dbl, const float* __restrict__ dt,
    const float* __restrict__ hin, const float* __restrict__ Dp,
    float* __restrict__ out) {
  const int wave = blockIdx.x * (blockDim.x >> 5) + (threadIdx.x >> 5);
  const int lane = threadIdx.x & 31;
  const int b     = wave / (NCHUNK * (D_INNER / 2));
  const int rem   = wave % (NCHUNK * (D_INNER / 2));
  const int chunk = rem / (D_INNER / 2);
  const int dpair = rem % (D_INNER / 2);
  const int d = dpair * 2 + (lane >> 4);
  const int n = lane & 15;

  const float A  = -__expf(A_log[d * STATE + n]);
  const float dp = Dp[d];
  const size_t sidx = (((size_t)b * D_INNER + d) * STATE + n) * NCHUNK + chunk;
  float h = hin[sidx];
  const int lbase = chunk * CLEN;
  for (int i = 0; i < CLEN; ++i) {
    const size_t r = (size_t)b * SEQ + (lbase + i);
    const float dtv = dt[r * D_INNER + d];
    const float uv  = u [r * D_INNER + d];
    const float Bv  = x_dbl[r * E_DIM + DT_RANK + n];
    const float Cv  = x_dbl[r * E_DIM + DT_RANK + STATE + n];
    const float a = __expf(dtv * A);
    h = fmaf(a, h, dtv * uv * Bv);
    float y = h * Cv;
    // reduce over 16 states (stays inside each half-wave group)
    y += __shfl_xor(y, 1, 32);
    y += __shfl_xor(y, 2, 32);
    y += __shfl_xor(y, 4, 32);
    y += __shfl_xor(y, 8, 32);
    if (n == 0) out[r * D_INNER + d] = fmaf(uv, dp, y);
  }
}

// ---------------------------------------------------------------------------
extern "C" void kernel_launch(void* const* d_in, const int* in_sizes, int n_in,
                              void* d_out, int out_size, void* d_ws, size_t ws_size,
                              hipStream_t stream) {
  const float* u     = (const float*)d_in[0];   // (2,2048,1024)
  const float* A_log = (const float*)d_in[1];   // (1024,16)
  const float* W_x   = (const float*)d_in[2];   // (96,1024)
  const float* W_dt  = (const float*)d_in[3];   // (1024,64)
  const float* b_dt  = (const float*)d_in[4];   // (1024,)
  const float* Dp    = (const float*)d_in[5];   // (1024,)
  float* out = (float*)d_out;

  float* ws    = (float*)d_ws;
  float* x_dbl = ws;                                   // 4096*96      = 393216
  float* dtbuf = x_dbl + (size_t)ROWS * E_DIM;         // 4096*1024    = 4194304
  float* prodA = dtbuf + (size_t)ROWS * D_INNER;       // 2*1024*16*8  = 262144
  float* hend  = prodA + (size_t)BATCH * D_INNER * STATE * NCHUNK;
  float* hin   = hend  + (size_t)BATCH * D_INNER * STATE * NCHUNK;

  // GEMM1: 256*6 = 1536 tiles, 4 waves/block
  xproj_wmma<<<1536 / 4, 128, 0, stream>>>(u, W_x, x_dbl);
  // GEMM2: 256*64 = 16384 tiles
  dtproj_wmma<<<16384 / 4, 128, 0, stream>>>(x_dbl, W_dt, b_dt, dtbuf);
  // Scan: 2*8*512 = 8192 waves, 8 waves/block
  scan_pass1<<<8192 / 8, 256, 0, stream>>>(u, A_log, x_dbl, dtbuf, prodA, hend);
  scan_combine<<<(BATCH * D_INNER * STATE) / 256, 256, 0, stream>>>(prodA, hend, hin);
  scan_pass3<<<8192 / 8, 256, 0, stream>>>(u, A_log, x_dbl, dtbuf, hin, Dp, out);
}